// GRU_2723009266180
// MI455X (gfx1250) — compile-verified
//
#include <hip/hip_runtime.h>

// GRU on MI455X (gfx1250): persistent fp32-WMMA recurrence.
//
// L=512, N=128, IN=256, H=512.  Grid: 32 blocks x 256 threads (8 wave32s).
//   block b  -> hidden-column tile  col0 = b*16
//   wave  w  -> batch-row tile      row0 = w*16
// All waves of a block share identical B fragments, so the block's W_hh
// slice (3 gates x 16 cols x 512 K = ~96 KB) is staged ONCE in LDS (320 KB
// per WGP on CDNA5) and reused for all 512 timesteps.
// The 512-step loop runs device-side; steps are separated by a software
// grid barrier (monotonic counter in d_ws, zeroed by an init kernel each
// launch -> graph-replay safe and deterministic).

typedef __attribute__((ext_vector_type(2))) float v2f;
typedef __attribute__((ext_vector_type(8))) float v8f;

constexpr int L_   = 512;
constexpr int N_   = 128;
constexpr int IN_  = 256;
constexpr int H_   = 512;
constexpr int NBLK = 32;                 // one block per column tile
constexpr int WSTRIDE = 3 * H_ + 4;      // padded LDS row stride (floats):
                                         // bank = 4*l15 + k (mod 64) -> conflict-free

__device__ __forceinline__ float sigm(float x) {
    return 1.0f / (1.0f + __expf(-x));
}
__device__ __forceinline__ float tanh_fast(float x) {
    return 2.0f / (1.0f + __expf(-2.0f * x)) - 1.0f;
}

__device__ __forceinline__ v8f wmma_f32(v2f a, v2f b, v8f c) {
    // D = A(16x4 f32) * B(4x16 f32) + C ; full-precision matrix pipe.
    return __builtin_amdgcn_wmma_f32_16x16x4_f32(
        /*neg_a=*/false, a, /*neg_b=*/false, b,
        /*c_mod=*/(short)0, c, /*reuse_a=*/false, /*reuse_b=*/false);
}

__global__ void init_counter_kernel(unsigned* __restrict__ cnt) {
    if (threadIdx.x == 0 && blockIdx.x == 0) *cnt = 0u;
}

__global__ __launch_bounds__(256)
void gru_persistent_kernel(const float* __restrict__ x,        // [L,N,IN]
                           float*       __restrict__ hbuf0,    // [N,H] ping
                           float*       __restrict__ hbuf1,    // [N,H] pong
                           const float* __restrict__ is_init,  // [L,N]
                           const float* __restrict__ Wih,      // [3H,IN]
                           const float* __restrict__ Whh,      // [3H,H]
                           const float* __restrict__ bih,      // [3H]
                           const float* __restrict__ bhh,      // [3H]
                           float*       __restrict__ out,      // [L,N,H]
                           unsigned*    __restrict__ bar)      // grid barrier
{
    __shared__ float ldsW[16][WSTRIDE];   // [col_local][g*512 + k], ~96.3 KB

    const int lane = threadIdx.x & 31;
    const int wave = threadIdx.x >> 5;
    const int col0 = blockIdx.x * 16;
    const int row0 = wave * 16;
    const int l15  = lane & 15;
    const int hi   = lane >> 4;           // A/B fragment halves: K=0,1 vs K=2,3
    const int arow = row0 + l15;          // A-fragment batch row
    const int c    = col0 + l15;          // output column of this lane

    // ---- stage W_hh column slice into LDS (once, reused 512x) -------------
    for (int i = threadIdx.x; i < 16 * 3 * H_; i += 256) {
        const int cc = i / (3 * H_);
        const int gk = i - cc * (3 * H_);         // g*512 + k
        const int g  = gk >> 9;
        const int k  = gk & (H_ - 1);
        ldsW[cc][gk] = Whh[((size_t)g * H_ + col0 + cc) * H_ + k];
    }
    __syncthreads();

    // per-lane column biases (time-invariant)
    const float bR  = bih[c]           + bhh[c];
    const float bZ  = bih[H_ + c]      + bhh[H_ + c];
    const float bXN = bih[2 * H_ + c];
    const float bHN = bhh[2 * H_ + c];

    // time-invariant B-fragment base pointers
    const float* ur = Wih + ((size_t)0 * H_ + c) * IN_ + 2 * hi;
    const float* uz = Wih + ((size_t)1 * H_ + c) * IN_ + 2 * hi;
    const float* un = Wih + ((size_t)2 * H_ + c) * IN_ + 2 * hi;
    const float* lr = &ldsW[l15][0 * H_ + 2 * hi];
    const float* lz = &ldsW[l15][1 * H_ + 2 * hi];
    const float* ln = &ldsW[l15][2 * H_ + 2 * hi];

    #pragma unroll 1
    for (int t = 0; t < L_; ++t) {
        const float* hp = (t & 1) ? hbuf1 : hbuf0;
        float*       hn = (t & 1) ? hbuf0 : hbuf1;

        // prefetch next timestep's x rows into cache while we compute
        if (t + 1 < L_)
            __builtin_prefetch(x + ((size_t)(t + 1) * N_ + arow) * IN_, 0, 3);

        const float mA = 1.0f - is_init[t * N_ + arow];

        v8f acc_r  = {0.f,0.f,0.f,0.f,0.f,0.f,0.f,0.f};
        v8f acc_z  = acc_r;
        v8f acc_xn = acc_r;
        v8f acc_hn = acc_r;

        // ---- hidden projection: K = 512, B fragments from LDS -------------
        {
            const float* ha = hp + (size_t)arow * H_ + 2 * hi;
            #pragma unroll 4
            for (int k = 0; k < H_; k += 4) {
                v2f a = *(const v2f*)(ha + k);
                a.x *= mA; a.y *= mA;
                v2f br = *(const v2f*)(lr + k);
                v2f bz = *(const v2f*)(lz + k);
                v2f bn = *(const v2f*)(ln + k);
                acc_r  = wmma_f32(a, br, acc_r);
                acc_z  = wmma_f32(a, bz, acc_z);
                acc_hn = wmma_f32(a, bn, acc_hn);
            }
        }

        // ---- input projection: K = 256, B fragments from global (L2) ------
        {
            const float* xa = x + ((size_t)t * N_ + arow) * IN_ + 2 * hi;
            #pragma unroll 4
            for (int k = 0; k < IN_; k += 4) {
                v2f a  = *(const v2f*)(xa + k);
                v2f br = *(const v2f*)(ur + k);
                v2f bz = *(const v2f*)(uz + k);
                v2f bn = *(const v2f*)(un + k);
                acc_r  = wmma_f32(a, br, acc_r);
                acc_z  = wmma_f32(a, bz, acc_z);
                acc_xn = wmma_f32(a, bn, acc_xn);
            }
        }

        // ---- gates + writeback --------------------------------------------
        // D layout: VGPR e holds row = row0 + e + 8*hi, col = col0 + l15.
        #pragma unroll
        for (int e = 0; e < 8; ++e) {
            const int   row = row0 + e + 8 * hi;
            const float m   = 1.0f - is_init[t * N_ + row];
            const float hm  = hp[(size_t)row * H_ + c] * m;
            const float r   = sigm(acc_r[e] + bR);
            const float z   = sigm(acc_z[e] + bZ);
            const float nn  = tanh_fast(acc_xn[e] + bXN + r * (acc_hn[e] + bHN));
            const float hv  = (1.0f - z) * nn + z * hm;
            hn[(size_t)row * H_ + c] = hv;
            out[((size_t)t * N_ + row) * H_ + c] = hv;
        }

        // ---- device-wide barrier (monotonic count, no reset needed) -------
        __threadfence();                       // release h_next stores
        __syncthreads();
        if (threadIdx.x == 0) {
            atomicAdd(bar, 1u);
            const unsigned tgt = (unsigned)NBLK * (unsigned)(t + 1);
            while (__hip_atomic_load(bar, __ATOMIC_RELAXED,
                                     __HIP_MEMORY_SCOPE_AGENT) < tgt)
                __builtin_amdgcn_s_sleep(1);
        }
        __syncthreads();
        __threadfence();                       // acquire others' h_next
    }
}

// Broadcast h_last over the L axis into the second output tensor [L,N,H].
__global__ __launch_bounds__(256)
void broadcast_last_kernel(const float* __restrict__ hlast, float* __restrict__ dst)
{
    const size_t i = (size_t)blockIdx.x * blockDim.x + threadIdx.x;
    dst[i] = hlast[i & ((size_t)N_ * H_ - 1)];
}

extern "C" void kernel_launch(void* const* d_in, const int* in_sizes, int n_in,
                              void* d_out, int out_size, void* d_ws, size_t ws_size,
                              hipStream_t stream)
{
    const float* input      = (const float*)d_in[0];  // [L,N,IN]
    const float* h0         = (const float*)d_in[1];  // [N,H]
    const float* is_initial = (const float*)d_in[2];  // [L,N,1]
    const float* Wih        = (const float*)d_in[3];  // [3H,IN]
    const float* Whh        = (const float*)d_in[4];  // [3H,H]
    const float* bih        = (const float*)d_in[5];  // [3H]
    const float* bhh        = (const float*)d_in[6];  // [3H]
    float* out = (float*)d_out;

    // workspace: [counter (256B-aligned slot)] [hbuf0][hbuf1]
    unsigned* bar = (unsigned*)d_ws;
    float* hbuf0 = (float*)((char*)d_ws + 256);
    float* hbuf1 = hbuf0 + (size_t)N_ * H_;

    init_counter_kernel<<<1, 64, 0, stream>>>(bar);
    hipMemcpyAsync(hbuf0, h0, (size_t)N_ * H_ * sizeof(float),
                   hipMemcpyDeviceToDevice, stream);

    gru_persistent_kernel<<<NBLK, 256, 0, stream>>>(
        input, hbuf0, hbuf1, is_initial, Wih, Whh, bih, bhh, out, bar);

    // L even -> final step (t=511, odd) wrote hbuf0
    const size_t total = (size_t)L_ * N_ * H_;
    broadcast_last_kernel<<<(unsigned)(total / 256), 256, 0, stream>>>(
        hbuf0, out + total);
}